// QualiCLIPLoss_19250043420909
// MI455X (gfx1250) — compile-verified
//
#include <hip/hip_runtime.h>
#include <math.h>

// ---------------------------------------------------------------------------
// QualiCLIP loss, MI455X (gfx1250).
//
// Phase 1 (bandwidth-bound, 32MB @ 23.3TB/s ~ 1.4us floor):
//   C(8192x16) = F(8192x1024) x B(1024x16), B col0=tp_hat, col1=tn_hat,
//   cols 2..15 zero-padded so the B operand load is branch-free.
//   V_WMMA_F32_16X16X4_F32, split-K=4 (2048 waves). Row sum-of-squares
//   runs on the VALU alongside (1 fma/elem), co-executing with the WMMA pipe.
// Phase 2: bucket the pairwise loss by pair_index (avg 8 rows/bucket ->
//   ~74K live pairs instead of 67M masked ones). Fully deterministic:
//   no float atomics; per-wave/per-block partials + fixed-order reductions.
// ---------------------------------------------------------------------------

#define NROW 8192
#define DCOL 1024
#define NUM_PAIRS 1024
#define MARGIN_RANK 0.05f
#define SPLITK 4
#define KCHUNK (DCOL / SPLITK)   // 256
#define NTILES (NROW / 16)       // 512
#define CAP 2048                 // max bucket members held in LDS

typedef __attribute__((ext_vector_type(2))) float v2f;
typedef __attribute__((ext_vector_type(8))) float v8f;

// Workspace layout (float offsets); total ~551KB.
#define OFF_B     0                        // 16*DCOL padded B matrix (64KB)
#define OFF_SPP   (OFF_B + 16*DCOL)        // SPLITK*NROW sim_pos partials
#define OFF_SNP   (OFF_SPP + SPLITK*NROW)  // SPLITK*NROW sim_neg partials
#define OFF_N2P   (OFF_SNP + SPLITK*NROW)  // SPLITK*NROW |x|^2 partials
#define OFF_SPF   (OFF_N2P + SPLITK*NROW)  // NROW final sim_pos
#define OFF_SNF   (OFF_SPF + NROW)         // NROW final sim_neg
#define OFF_BSUM  (OFF_SNF + NROW)         // NUM_PAIRS*3 floats
#define OFF_BCNT  (OFF_BSUM + NUM_PAIRS*3) // NUM_PAIRS*2 uints

// ---- normalize text vectors and build the padded 1024x16 B matrix --------
__global__ __launch_bounds__(256) void textnorm_kernel(
    const float* __restrict__ tp, const float* __restrict__ tn,
    float* __restrict__ Bbuf)
{
  __shared__ float sA[256], sB[256];
  const int tid = threadIdx.x;
  float a = 0.f, b = 0.f;
  for (int i = tid; i < DCOL; i += 256) {
    float x = tp[i]; a = fmaf(x, x, a);
    float y = tn[i]; b = fmaf(y, y, b);
  }
  sA[tid] = a; sB[tid] = b;
  for (int st = 128; st; st >>= 1) {
    __syncthreads();
    if (tid < st) { sA[tid] += sA[tid + st]; sB[tid] += sB[tid + st]; }
  }
  __syncthreads();
  const float ra = 1.f / sqrtf(sA[0]);
  const float rb = 1.f / sqrtf(sB[0]);
  // Bbuf[n][k]: n==0 -> tp_hat, n==1 -> tn_hat, n>=2 -> 0
  for (int i = tid; i < 16 * DCOL; i += 256) {
    const int n = i >> 10, k = i & (DCOL - 1);
    float v = 0.f;
    if (n == 0) v = tp[k] * ra;
    else if (n == 1) v = tn[k] * rb;
    Bbuf[i] = v;
  }
}

// ---- skinny GEMM on the matrix pipe, split-K -----------------------------
// One wave = one (16-row tile, K-chunk). A layout (f32 16x4): lane l holds
// row M=l&15, K = 2*(l>>4)+{0,1} -> b64 load. B layout mirrors with N=l&15.
// B loads are unconditional (zero-padded table) -> branch-free inner loop.
__global__ __launch_bounds__(256) void gemm_wmma_kernel(
    const float* __restrict__ F, const float* __restrict__ Bbuf,
    float* __restrict__ sp_part, float* __restrict__ sn_part,
    float* __restrict__ n2_part)
{
  const int lane = threadIdx.x & 31;
  const int w    = (blockIdx.x << 3) + (threadIdx.x >> 5);
  const int tile = w >> 2;      // 0..511
  const int s    = w & 3;       // split-K chunk
  if (tile >= NTILES) return;   // wave-uniform: EXEC stays all-ones for WMMA

  const int m = lane & 15;      // row within tile (A) / column index (B)
  const int h = lane >> 4;      // K-pair selector
  const float* rowp = F    + (size_t)(tile * 16 + m) * DCOL + s * KCHUNK + 2 * h;
  const float* bp   = Bbuf + (size_t)m * DCOL              + s * KCHUNK + 2 * h;

  v8f c = {0.f, 0.f, 0.f, 0.f, 0.f, 0.f, 0.f, 0.f};
  float nacc = 0.f;
  #pragma unroll 8
  for (int kb = 0; kb < KCHUNK; kb += 4) {
    v2f a  = *(const v2f*)(rowp + kb);  // global_load_b64, streams row
    v2f bv = *(const v2f*)(bp + kb);    // hot 64KB table (L1-resident)
    nacc = fmaf(a.x, a.x, nacc);
    nacc = fmaf(a.y, a.y, nacc);
    // (neg_a, A, neg_b, B, c_mod, C, reuse_a, reuse_b)
    c = __builtin_amdgcn_wmma_f32_16x16x4_f32(false, a, false, bv,
                                              (short)0, c, false, false);
  }
  // lane and lane^16 cover complementary K mod-4 halves of the row chunk
  nacc += __shfl_xor(nacc, 16, 32);
  if (lane < 16) n2_part[(size_t)s * NROW + tile * 16 + m] = nacc;

  // C layout: vgpr r, lanes<16 -> (M=r, N=lane); lanes>=16 -> (M=r+8, N=lane-16)
  if (m < 2) {
    float* dst = (m == 0) ? sp_part : sn_part;
    const int base = s * NROW + tile * 16 + h * 8;
    #pragma unroll
    for (int r = 0; r < 8; ++r) dst[base + r] = c[r];
  }
}

// ---- combine split-K partials, apply row rsqrt ---------------------------
__global__ __launch_bounds__(256) void scale_kernel(
    const float* __restrict__ sp_part, const float* __restrict__ sn_part,
    const float* __restrict__ n2_part, float* __restrict__ spf,
    float* __restrict__ snf)
{
  const int i = blockIdx.x * 256 + threadIdx.x;
  if (i >= NROW) return;
  float sp = 0.f, sn = 0.f, n2 = 0.f;
  #pragma unroll
  for (int s = 0; s < SPLITK; ++s) {
    sp += sp_part[s * NROW + i];
    sn += sn_part[s * NROW + i];
    n2 += n2_part[s * NROW + i];
  }
  const float r = 1.f / sqrtf(n2);
  spf[i] = sp * r; snf[i] = sn * r;
}

// ---- per-bucket pairwise losses ------------------------------------------
// The upper-triangle mask just means "count each unordered pair once";
// cons terms are symmetric, rank terms pick the (lev_lo, lev_hi) orientation,
// so original indices are irrelevant.
__global__ __launch_bounds__(256) void bucket_kernel(
    const int* __restrict__ pair_idx, const int* __restrict__ levels,
    const float* __restrict__ sp, const float* __restrict__ sn,
    float* __restrict__ bsum, unsigned* __restrict__ bcnt)
{
  __shared__ float s_sp[CAP], s_sn[CAP];
  __shared__ int   s_lev[CAP];
  __shared__ int   s_scan[257];
  __shared__ float redf[3][8];
  __shared__ unsigned redu[2][8];

  const int b = blockIdx.x, tid = threadIdx.x;

  // deterministic compaction: count, serial prefix scan, scatter
  int cnt = 0;
  for (int i = tid; i < NROW; i += 256) cnt += (pair_idx[i] == b);
  s_scan[tid + 1] = cnt;
  __syncthreads();
  if (tid == 0) {
    s_scan[0] = 0;
    for (int t = 1; t <= 256; ++t) s_scan[t] += s_scan[t - 1];
  }
  __syncthreads();
  int pos = s_scan[tid];
  for (int i = tid; i < NROW; i += 256) {
    if (pair_idx[i] == b) {
      if (pos < CAP) { s_sp[pos] = sp[i]; s_sn[pos] = sn[i]; s_lev[pos] = levels[i]; }
      ++pos;
    }
  }
  __syncthreads();
  const int n = min(s_scan[256], CAP);

  float csum = 0.f, psum = 0.f, qsum = 0.f;
  unsigned ccnt = 0u, rcnt = 0u;
  for (int t = tid; t < n * n; t += 256) {
    const int ia = t / n, ib = t % n;
    if (ib <= ia) continue;                    // unordered pairs
    const int la = s_lev[ia], lb = s_lev[ib];
    const float spa = s_sp[ia], spb = s_sp[ib];
    const float sna = s_sn[ia], snb = s_sn[ib];
    if (la == lb) {
      csum += fabsf(spa - spb) + fabsf(sna - snb);
      ++ccnt;
    } else {
      float sp_lo, sp_hi, sn_lo, sn_hi;        // i = lower level, j = higher
      if (la < lb) { sp_lo = spa; sp_hi = spb; sn_lo = sna; sn_hi = snb; }
      else         { sp_lo = spb; sp_hi = spa; sn_lo = snb; sn_hi = sna; }
      psum += fmaxf(0.f, MARGIN_RANK - (sp_lo - sp_hi));
      qsum += fmaxf(0.f, MARGIN_RANK + (sn_lo - sn_hi));
      ++rcnt;
    }
  }
  // wave32 reduction (fixed order), then fixed-order cross-wave sum
  for (int off = 16; off; off >>= 1) {
    csum += __shfl_down(csum, off, 32);
    psum += __shfl_down(psum, off, 32);
    qsum += __shfl_down(qsum, off, 32);
    ccnt += __shfl_down(ccnt, off, 32);
    rcnt += __shfl_down(rcnt, off, 32);
  }
  const int wv = tid >> 5;
  if ((tid & 31) == 0) {
    redf[0][wv] = csum; redf[1][wv] = psum; redf[2][wv] = qsum;
    redu[0][wv] = ccnt; redu[1][wv] = rcnt;
  }
  __syncthreads();
  if (tid == 0) {
    float c0 = 0.f, p0 = 0.f, q0 = 0.f; unsigned u0 = 0u, u1 = 0u;
    for (int i = 0; i < 8; ++i) {
      c0 += redf[0][i]; p0 += redf[1][i]; q0 += redf[2][i];
      u0 += redu[0][i]; u1 += redu[1][i];
    }
    bsum[b * 3 + 0] = c0; bsum[b * 3 + 1] = p0; bsum[b * 3 + 2] = q0;
    bcnt[b * 2 + 0] = u0; bcnt[b * 2 + 1] = u1;
  }
}

// ---- final reduction over 1024 buckets -----------------------------------
__global__ __launch_bounds__(256) void finalize_kernel(
    const float* __restrict__ bsum, const unsigned* __restrict__ bcnt,
    float* __restrict__ out)
{
  __shared__ float sc[256], sp_[256], sq[256];
  __shared__ unsigned su[256], sv[256];
  const int tid = threadIdx.x;
  float c = 0.f, p = 0.f, q = 0.f; unsigned u = 0u, v = 0u;
  for (int b = tid; b < NUM_PAIRS; b += 256) {
    c += bsum[b * 3 + 0]; p += bsum[b * 3 + 1]; q += bsum[b * 3 + 2];
    u += bcnt[b * 2 + 0]; v += bcnt[b * 2 + 1];
  }
  sc[tid] = c; sp_[tid] = p; sq[tid] = q; su[tid] = u; sv[tid] = v;
  for (int st = 128; st; st >>= 1) {
    __syncthreads();
    if (tid < st) {
      sc[tid] += sc[tid + st]; sp_[tid] += sp_[tid + st]; sq[tid] += sq[tid + st];
      su[tid] += su[tid + st]; sv[tid] += sv[tid + st];
    }
  }
  __syncthreads();
  if (tid == 0) {
    const unsigned ccnt = su[0], rcnt = sv[0];
    const float loss_cons = (ccnt > 0u) ? sc[0] / (float)(2u * ccnt) : 0.f;
    const float loss_pos  = (rcnt > 0u) ? sp_[0] / (float)rcnt : 0.f;
    const float loss_neg  = (rcnt > 0u) ? sq[0] / (float)rcnt : 0.f;
    out[0] = loss_cons + loss_pos + loss_neg;  // lambdas are all 1.0
  }
}

// ---------------------------------------------------------------------------
extern "C" void kernel_launch(void* const* d_in, const int* in_sizes, int n_in,
                              void* d_out, int out_size, void* d_ws, size_t ws_size,
                              hipStream_t stream) {
  (void)in_sizes; (void)n_in; (void)out_size; (void)ws_size;
  const float* F      = (const float*)d_in[0];   // (8192,1024)
  const float* tp     = (const float*)d_in[1];   // (1024,)
  const float* tn     = (const float*)d_in[2];   // (1024,)
  const int*   levels = (const int*)d_in[3];     // (8192,)
  const int*   pidx   = (const int*)d_in[4];     // (8192,)
  float* out = (float*)d_out;

  float* ws = (float*)d_ws;                      // needs ~551KB
  float* Bbuf    = ws + OFF_B;
  float* sp_part = ws + OFF_SPP;
  float* sn_part = ws + OFF_SNP;
  float* n2_part = ws + OFF_N2P;
  float* spf     = ws + OFF_SPF;
  float* snf     = ws + OFF_SNF;
  float* bsum    = ws + OFF_BSUM;
  unsigned* bcnt = (unsigned*)(ws + OFF_BCNT);

  textnorm_kernel<<<1, 256, 0, stream>>>(tp, tn, Bbuf);
  // 512 tiles x split-K 4 = 2048 waves = 256 blocks of 8 waves
  gemm_wmma_kernel<<<256, 256, 0, stream>>>(F, Bbuf, sp_part, sn_part, n2_part);
  scale_kernel<<<NROW / 256, 256, 0, stream>>>(sp_part, sn_part, n2_part, spf, snf);
  bucket_kernel<<<NUM_PAIRS, 256, 0, stream>>>(pidx, levels, spf, snf, bsum, bcnt);
  finalize_kernel<<<1, 256, 0, stream>>>(bsum, bcnt, out);
}